// S4Conv1D_59459527246267
// MI455X (gfx1250) — compile-verified
//
#include <hip/hip_runtime.h>
#include <math.h>

// ---------------------------------------------------------------------------
// S4 convolution for MI455X (gfx1250, wave32, WMMA).
//   prologue : diag, power-iteration SVD (rank 1), triangular solve for Ad,
//              Ad^(3^11) via 11 cubings with V_WMMA_F32_16X16X4_F32 GEMMs
//   main     : Cauchy kernel as an on-the-fly f32 WMMA GEMM [L x 512]x[512 x 16]
//   FFTs     : radix-3 Stockham (L = 3^11), 2 transforms (identity removes 3rd)
// Everything (<= ~5 MB) stays resident in the 192 MB L2.
// ---------------------------------------------------------------------------

#define NST   256
#define LSEQ  177147          // 3^11
#define L3RD  59049           // LSEQ / 3
#define PI_D  3.14159265358979323846

typedef __attribute__((ext_vector_type(2))) float v2f;
typedef __attribute__((ext_vector_type(8))) float v8f;

__device__ __forceinline__ v8f wmma_f32_4(v2f a, v2f b, v8f c) {
  // V_WMMA_F32_16X16X4_F32 : D = A(16x4) * B(4x16) + C(16x16), all f32
  return __builtin_amdgcn_wmma_f32_16x16x4_f32(
      /*neg_a=*/false, a, /*neg_b=*/false, b,
      /*c_mod=*/(short)0, c, /*reuse_a=*/false, /*reuse_b=*/false);
}

// ---------------------------------------------------------------------------
// Ad = (I - 0.5 A)^-1 (I + 0.5 A) by forward substitution (A lower-triangular).
// One block, thread j owns column j (private column history -> scratch).
// ---------------------------------------------------------------------------
__global__ void prep_tri(const float* __restrict__ A, float* __restrict__ Ad) {
  __shared__ float rowA[NST];
  int j = threadIdx.x;
  float col[NST];
  for (int i = 0; i < NST; ++i) {
    __syncthreads();
    rowA[j] = A[i * NST + j];
    __syncthreads();
    float s = 0.f;
    for (int k = 0; k < i; ++k) s += (-0.5f * rowA[k]) * col[k];
    float rhs = 0.5f * rowA[j] + (i == j ? 1.f : 0.f);
    float tii = 1.f - 0.5f * rowA[i];
    col[i] = (rhs - s) / tii;
  }
  for (int i = 0; i < NST; ++i) Ad[i * NST + j] = col[i];
}

// ---------------------------------------------------------------------------
// 256x256x256 f32 GEMM, one wave per 16x16 tile, V_WMMA_F32_16X16X4_F32.
// A frag: lane(0..15)=row M, VGPR g holds K = k0 + g (+2 for lanes 16..31).
// B frag: VGPR g holds row K = k0 + g (+2 for hi half), col = lane & 15.
// ---------------------------------------------------------------------------
__global__ void mm256_wmma(const float* __restrict__ A,
                           const float* __restrict__ B,
                           float* __restrict__ C) {
  int lane = threadIdx.x;
  int tm = blockIdx.y * 16, tn = blockIdx.x * 16;
  int row  = tm + (lane & 15);
  int colb = tn + (lane & 15);
  int khi  = (lane >> 4) << 1;          // 0 or 2
  v8f acc = {0.f, 0.f, 0.f, 0.f, 0.f, 0.f, 0.f, 0.f};
  for (int k0 = 0; k0 < NST; k0 += 4) {
    int k = k0 + khi;
    v2f a; a.x = A[row * NST + k];       a.y = A[row * NST + k + 1];
    v2f b; b.x = B[k * NST + colb];      b.y = B[(k + 1) * NST + colb];
    acc = wmma_f32_4(a, b, acc);
  }
  int rbase = tm + ((lane >> 4) << 3);   // C: VGPR r -> row r (lo) / r+8 (hi)
  for (int r = 0; r < 8; ++r) C[(rbase + r) * NST + colb] = acc[r];
}

// ---------------------------------------------------------------------------
// lam, rank-1 SVD by power iteration, cts = C - AdL^T C, weight vectors.
// All quantities real (M real => singular vectors real; sign cancels).
// ---------------------------------------------------------------------------
__global__ void prep2(const float* __restrict__ A, const float* __restrict__ Bv,
                      const float* __restrict__ Cv, const float* __restrict__ AdL,
                      float* __restrict__ lamOut, float* __restrict__ Wout) {
  __shared__ float u[NST], v[NST], red[NST];
  int t = threadIdx.x;
  float lam = A[t * NST + t];
  lamOut[t] = lam;
  v[t] = 1.0f;
  __syncthreads();
  float sigma = 0.f;
  for (int it = 0; it < 64; ++it) {
    float s = 0.f;                               // u = M v   (M strictly lower)
    for (int j = 0; j < NST; ++j) if (j < t) s += A[t * NST + j] * v[j];
    u[t] = s; red[t] = s * s; __syncthreads();
    for (int off = 128; off > 0; off >>= 1) {
      if (t < off) red[t] += red[t + off];
      __syncthreads();
    }
    float nu = sqrtf(red[0]); __syncthreads();
    u[t] = (nu > 0.f) ? u[t] / nu : 0.f; __syncthreads();
    float q = 0.f;                               // q = M^T u
    for (int i = t + 1; i < NST; ++i) q += A[i * NST + t] * u[i];
    v[t] = q; red[t] = q * q; __syncthreads();
    for (int off = 128; off > 0; off >>= 1) {
      if (t < off) red[t] += red[t + off];
      __syncthreads();
    }
    sigma = sqrtf(red[0]); __syncthreads();
    v[t] = (sigma > 0.f) ? v[t] / sigma : 0.f; __syncthreads();
  }
  float sq = sqrtf(sigma);
  float Pn = u[t] * sq, Qn = v[t] * sq;
  float s2 = 0.f;                                // cts = C - AdL^T C
  for (int j = 0; j < NST; ++j) s2 += AdL[j * NST + t] * Cv[j];
  float cts = Cv[t] - s2;
  float b = Bv[t];
  Wout[0 * NST + t] = cts * b;                   // k00 weight
  Wout[1 * NST + t] = cts * Pn;                  // k01 weight
  Wout[2 * NST + t] = Qn * b;                    // k10 weight
  Wout[3 * NST + t] = Qn * Pn;                   // k11 weight
}

// ---------------------------------------------------------------------------
// Cauchy kernel as WMMA GEMM. Wave handles 16 consecutive l values.
// K dim = 2N (re/im interleaved resolvent), generated in-register per step:
//   r(l,n) = (-lam[n] - i*gi(l)) / (lam[n]^2 + gi(l)^2),  gi = -2 tan(pi*l/L)
// B (in LDS): B[2n+c][2m+cc] = w_m[n] iff c==cc, else 0  (weights real).
// Cols 0..7 of the 16x16 accumulator = re/im of k00,k01,k10,k11.
// ---------------------------------------------------------------------------
__global__ void cauchy_wmma(const float* __restrict__ lamG,
                            const float* __restrict__ W,
                            float2* __restrict__ Kc) {
  __shared__ float2 lamLds[NST];        // (-lam, lam^2)
  __shared__ float2 bLds[NST * 16];     // [n][col] = {B[2n][col], B[2n+1][col]}
  __shared__ float  cSpill[8 * 16 * 16];
  int tid = threadIdx.x;
  {
    float lm = lamG[tid];
    lamLds[tid] = make_float2(-lm, lm * lm);
  }
  for (int e = tid; e < NST * 16; e += 256) {
    int n = e >> 4, colv = e & 15;
    float b0 = 0.f, b1 = 0.f;
    if (colv < 8) {
      float w = W[(colv >> 1) * NST + n];
      if ((colv & 1) == 0) b0 = w; else b1 = w;
    }
    bLds[e] = make_float2(b0, b1);
  }
  __syncthreads();

  int wave = tid >> 5, lane = tid & 31;
  int tile = blockIdx.x * 8 + wave;
  int l0 = tile * 16;
  int rowi = lane & 15;
  int l = l0 + rowi;
  double th2 = PI_D * (double)l / (double)LSEQ;
  float gi  = (float)(-2.0 * tan(th2));          // g = -2i*tan(theta/2)
  float gi2 = gi * gi;
  float gin = -gi;
  int hi = lane >> 4;
  int colIdx = lane & 15;

  v8f acc = {0.f, 0.f, 0.f, 0.f, 0.f, 0.f, 0.f, 0.f};
#pragma unroll 4
  for (int it = 0; it < 128; ++it) {
    int n = it * 2 + hi;
    float2 lp = lamLds[n];
    float2 bp = bLds[n * 16 + colIdx];
    float inv = __builtin_amdgcn_rcpf(lp.y + gi2);
    v2f a; a.x = lp.x * inv; a.y = gin * inv;     // (rr, ri)
    v2f b; b.x = bp.x;       b.y = bp.y;
    acc = wmma_f32_4(a, b, acc);
  }

  float* cs = &cSpill[wave * 256];
  for (int r = 0; r < 8; ++r) cs[(hi * 8 + r) * 16 + colIdx] = acc[r];
  __syncthreads();

  if (lane < 16) {
    int ll = l0 + lane;
    if (ll < LSEQ) {
      float* row = &cs[lane * 16];
      float k00r = row[0], k00i = row[1], k01r = row[2], k01i = row[3];
      float k10r = row[4], k10i = row[5], k11r = row[6], k11i = row[7];
      float dr = 1.f + k11r, di = k11i;          // t = k10 / (1 + k11)
      float dinv = 1.f / (dr * dr + di * di);
      float tr = (k10r * dr + k10i * di) * dinv;
      float ti = (k10i * dr - k10r * di) * dinv;
      float nr = k00r - (k01r * tr - k01i * ti);
      float ni = k00i - (k01r * ti + k01i * tr);
      double th = 2.0 * PI_D * (double)ll / (double)LSEQ;
      double zr = cos(th), zi = sin(th);
      double pr = 1.0 + zr, pim = zi;            // w = 2/(1+z)
      double pinv = 2.0 / (pr * pr + pim * pim);
      float wr = (float)(pr * pinv), wi = (float)(-pim * pinv);
      Kc[ll] = make_float2(nr * wr - ni * wi, nr * wi + ni * wr);
    }
  }
}

// ---------------------------------------------------------------------------
// FFT: radix-3 Stockham, forward sign. L = 3^11 -> 11 passes.
// ---------------------------------------------------------------------------
__global__ void expand_real(const float* __restrict__ x, float2* __restrict__ o) {
  int j = blockIdx.x * 256 + threadIdx.x;
  if (j < LSEQ) o[j] = make_float2(x[j], 0.f);
}

__global__ void fft3_pass(const float2* __restrict__ in, float2* __restrict__ out,
                          int Ns) {
  int j = blockIdx.x * 256 + threadIdx.x;
  if (j >= L3RD) return;
  int idx = j % Ns;
  int blk = j / Ns;
  float2 a = in[j], b = in[j + L3RD], c = in[j + 2 * L3RD];
  double ang = -2.0 * PI_D * (double)idx / (double)(3 * Ns);
  double w1r = cos(ang), w1i = sin(ang);
  double w2r = w1r * w1r - w1i * w1i, w2i = 2.0 * w1r * w1i;
  float br = (float)(b.x * w1r - b.y * w1i), bi = (float)(b.x * w1i + b.y * w1r);
  float cr = (float)(c.x * w2r - c.y * w2i), ci = (float)(c.x * w2i + c.y * w2r);
  const float s3 = 0.8660254037844386f;
  float mr = a.x - 0.5f * (br + cr), mi = a.y - 0.5f * (bi + ci);
  float dr = s3 * (br - cr),         di = s3 * (bi - ci);
  int ob = blk * (3 * Ns) + idx;
  out[ob]          = make_float2(a.x + br + cr, a.y + bi + ci);
  out[ob + Ns]     = make_float2(mr + di, mi - dr);
  out[ob + 2 * Ns] = make_float2(mr - di, mi + dr);
}

// Y[j] = conj( X[j] * H[j] ),  H[j] = (Kc[(L-j)%L] + conj(Kc[j])) / 2
// (identity replaces the reference's fft(Kd); conj feeds the inverse FFT trick)
__global__ void pointwise(const float2* __restrict__ X,
                          const float2* __restrict__ Kc,
                          float2* __restrict__ out) {
  int j = blockIdx.x * 256 + threadIdx.x;
  if (j >= LSEQ) return;
  float2 kj = Kc[j];
  float2 km = Kc[(j == 0) ? 0 : (LSEQ - j)];
  float hr = 0.5f * (km.x + kj.x);
  float hi = 0.5f * (km.y - kj.y);
  float2 Xj = X[j];
  float pr = Xj.x * hr - Xj.y * hi;
  float pi = Xj.x * hi + Xj.y * hr;
  out[j] = make_float2(pr, -pi);
}

__global__ void finalize(const float2* __restrict__ in, float* __restrict__ y) {
  int j = blockIdx.x * 256 + threadIdx.x;
  if (j < LSEQ) y[j] = in[j].x * (1.0f / (float)LSEQ);
}

// ---------------------------------------------------------------------------
extern "C" void kernel_launch(void* const* d_in, const int* in_sizes, int n_in,
                              void* d_out, int out_size, void* d_ws, size_t ws_size,
                              hipStream_t stream) {
  (void)in_sizes; (void)n_in; (void)out_size; (void)ws_size;
  const float* x  = (const float*)d_in[0];
  const float* A  = (const float*)d_in[1];
  const float* Bv = (const float*)d_in[2];
  const float* Cv = (const float*)d_in[3];
  float* out = (float*)d_out;
  char* ws = (char*)d_ws;

  float*  Ad0  = (float*)(ws + 0);
  float*  Ad1  = (float*)(ws + 262144);
  float*  AdT  = (float*)(ws + 524288);
  float*  lam  = (float*)(ws + 786432);
  float*  W    = (float*)(ws + 790528);
  float2* Kc   = (float2*)(ws + 798720);
  float2* bufA = (float2*)(ws + 2215936);
  float2* bufB = (float2*)(ws + 3633152);

  // --- prologue: Ad, Ad^(3^11) via 11 cubings (22 WMMA GEMMs), weights ---
  prep_tri<<<1, 256, 0, stream>>>(A, Ad0);
  float* Xc = Ad0; float* Xo = Ad1;
  for (int it = 0; it < 11; ++it) {
    mm256_wmma<<<dim3(16, 16), 32, 0, stream>>>(Xc, Xc, AdT);
    mm256_wmma<<<dim3(16, 16), 32, 0, stream>>>(AdT, Xc, Xo);
    float* t = Xc; Xc = Xo; Xo = t;
  }
  prep2<<<1, 256, 0, stream>>>(A, Bv, Cv, Xc, lam, W);

  // --- Cauchy kernel: 11072 l-tiles of 16, 8 waves/block ---
  cauchy_wmma<<<1384, 256, 0, stream>>>(lam, W, Kc);

  // --- forward FFT of x ---
  expand_real<<<(LSEQ + 255) / 256, 256, 0, stream>>>(x, bufA);
  float2* pA = bufA; float2* pB = bufB;
  int Ns = 1;
  for (int p = 0; p < 11; ++p) {
    fft3_pass<<<(L3RD + 255) / 256, 256, 0, stream>>>(pA, pB, Ns);
    Ns *= 3;
    float2* t = pA; pA = pB; pB = t;
  }
  // X lives in pA

  // --- multiply by H(Kc), conj, inverse FFT via forward transform ---
  pointwise<<<(LSEQ + 255) / 256, 256, 0, stream>>>(pA, Kc, pB);
  float2* qA = pB; float2* qB = pA;
  Ns = 1;
  for (int p = 0; p < 11; ++p) {
    fft3_pass<<<(L3RD + 255) / 256, 256, 0, stream>>>(qA, qB, Ns);
    Ns *= 3;
    float2* t = qA; qA = qB; qB = t;
  }
  finalize<<<(LSEQ + 255) / 256, 256, 0, stream>>>(qA, out);
}